// LoKrModule_86646670229807
// MI455X (gfx1250) — compile-verified
//
#include <hip/hip_runtime.h>

typedef __attribute__((ext_vector_type(2))) float v2f;
typedef __attribute__((ext_vector_type(8))) float v8f;
typedef __attribute__((ext_vector_type(4))) int   v4i;

#define AS1 __attribute__((address_space(1)))
#define AS3 __attribute__((address_space(3)))

#define IN_DIM  4096
#define OUT_DIM 4096
#define M_TOT   16384   // B*S = 4*4096
#define BM 128
#define BN 128
#define BK 32
#define LDA 36          // padded stride: conflict-free fragment reads, 16B-aligned f4 stores
#define LDB 36
#define SCALE 0.25f     // alpha/lora_dim * multiplier = 1/4

#if __has_builtin(__builtin_amdgcn_global_load_async_to_lds_b128)
#define HAVE_ASYNC_LDS 1
#else
#define HAVE_ASYNC_LDS 0
#endif

__global__ __launch_bounds__(256)
void lokr_fused_wmma_gemm(const float* __restrict__ x,
                          const float* __restrict__ W0,
                          const float* __restrict__ bias,
                          const float* __restrict__ w1,
                          const float* __restrict__ w2a,
                          const float* __restrict__ w2b,
                          float* __restrict__ out)
{
    __shared__ float ldsA[2][BM * LDA];
    __shared__ float ldsB[2][BN * LDB];
    __shared__ float ldsW2[64 * 64];

    const int tid  = threadIdx.x;
    const int lane = tid & 31;          // wave32
    const int wave = tid >> 5;          // 8 waves
    const int lsub = lane & 15;
    const int lhi  = lane >> 4;         // 0 or 1

    const int n0 = blockIdx.x * BN;
    const int m0 = blockIdx.y * BM;

    // ---- w2 = w2a(64x4) @ w2b(4x64), staged once into LDS ----
    for (int idx = tid; idx < 64 * 64; idx += 256) {
        const int o  = idx >> 6;
        const int ii = idx & 63;
        float s = 0.f;
#pragma unroll
        for (int r = 0; r < 4; ++r)
            s = fmaf(w2a[o * 4 + r], w2b[r * 64 + ii], s);
        ldsW2[idx] = s;
    }
    __syncthreads();

    // ---- A tile staging: 128x32 f32, async direct-to-LDS when available ----
    auto stageA = [&](int k0, int buf) {
#pragma unroll
        for (int jj = 0; jj < 4; ++jj) {
            const int idx  = tid + jj * 256;     // 1024 float4 slots
            const int row  = idx >> 3;           // 8 float4 per row
            const int col4 = (idx & 7) << 2;
            const float* gsrc = x + (size_t)(m0 + row) * IN_DIM + k0 + col4;
            float* ldst = &ldsA[buf][row * LDA + col4];
#if HAVE_ASYNC_LDS
            __builtin_amdgcn_global_load_async_to_lds_b128(
                (AS1 v4i*)const_cast<float*>(gsrc), (AS3 v4i*)ldst, 0, 0);
#else
            *(float4*)ldst = *(const float4*)gsrc;
#endif
        }
    };

    // ---- B tile: global->regs (issue), then fused LoKr FMA + ds_store ----
    auto loadB = [&](int k0, float4 w0v[4], float4 w2vv[4], float w1v[4]) {
        const int kdiv64 = k0 >> 6;
        const int kmod   = k0 & 63;
#pragma unroll
        for (int jj = 0; jj < 4; ++jj) {
            const int idx  = tid + jj * 256;
            const int row  = idx >> 3;
            const int col4 = (idx & 7) << 2;
            const int n    = n0 + row;
            w1v[jj]  = SCALE * w1[(n >> 6) * 64 + kdiv64];
            w0v[jj]  = *(const float4*)(W0 + (size_t)n * IN_DIM + k0 + col4);
            w2vv[jj] = *(const float4*)(&ldsW2[(n & 63) * 64 + kmod + col4]);
        }
    };
    auto storeB = [&](int buf, const float4 w0v[4], const float4 w2vv[4],
                      const float w1v[4]) {
#pragma unroll
        for (int jj = 0; jj < 4; ++jj) {
            const int idx  = tid + jj * 256;
            const int row  = idx >> 3;
            const int col4 = (idx & 7) << 2;
            float4 bv;
            bv.x = fmaf(w1v[jj], w2vv[jj].x, w0v[jj].x);
            bv.y = fmaf(w1v[jj], w2vv[jj].y, w0v[jj].y);
            bv.z = fmaf(w1v[jj], w2vv[jj].z, w0v[jj].z);
            bv.w = fmaf(w1v[jj], w2vv[jj].w, w0v[jj].w);
            *(float4*)(&ldsB[buf][row * LDB + col4]) = bv;
        }
    };

    // wave sub-tile: 64 (M) x 32 (N); 2x4 wave grid covers 128x128
    const int mw = (wave >> 2) * 64;    // 0 or 64
    const int nw = (wave & 3) * 32;     // 0,32,64,96

    v8f acc[4][2];
    const v8f vzero = {};
#pragma unroll
    for (int i = 0; i < 4; ++i)
#pragma unroll
        for (int j = 0; j < 2; ++j)
            acc[i][j] = vzero;

    // ---- prologue: stage k0 = 0 into buffer 0 ----
    {
        stageA(0, 0);
        float4 w0v[4], w2vv[4];
        float  w1v[4];
        loadB(0, w0v, w2vv, w1v);
        storeB(0, w0v, w2vv, w1v);
#if HAVE_ASYNC_LDS
        __builtin_amdgcn_s_wait_asynccnt(0);
#endif
        __syncthreads();
    }

    // ---- main loop: double-buffered, compute(buf) overlapped with stage(nxt)
    for (int k0 = 0, it = 0; k0 < IN_DIM; k0 += BK, ++it) {
        const int buf = it & 1;
        const int nxt = buf ^ 1;
        const int k1  = k0 + BK;
        const bool more = (k1 < IN_DIM);

        float4 w0v[4], w2vv[4];
        float  w1v[4];
        if (more) {
            stageA(k1, nxt);                 // async: no VGPR round-trip
            loadB(k1, w0v, w2vv, w1v);       // global loads in flight over compute
        }

        // ---- compute: 8 k-substeps x 8 WMMAs on buffer `buf` ----
#pragma unroll
        for (int kk = 0; kk < BK; kk += 4) {
            v2f afr[4], bfr[2];
            // A 16x4 f32 layout: lanes 0-15 -> K=k..k+1, lanes 16-31 -> K=k+2..k+3
#pragma unroll
            for (int i = 0; i < 4; ++i)
                afr[i] = *(const v2f*)(&ldsA[buf][(mw + 16 * i + lsub) * LDA + kk + 2 * lhi]);
#pragma unroll
            for (int j = 0; j < 2; ++j)
                bfr[j] = *(const v2f*)(&ldsB[buf][(nw + 16 * j + lsub) * LDB + kk + 2 * lhi]);
#pragma unroll
            for (int i = 0; i < 4; ++i)
#pragma unroll
                for (int j = 0; j < 2; ++j)
                    acc[i][j] = __builtin_amdgcn_wmma_f32_16x16x4_f32(
                        false, afr[i], false, bfr[j],
                        (short)0, acc[i][j], false, false);
        }

        if (more) {
            storeB(nxt, w0v, w2vv, w1v);     // fused LoKr delta FMA into LDS
#if HAVE_ASYNC_LDS
            __builtin_amdgcn_s_wait_asynccnt(0);
#endif
        }
        __syncthreads();
    }

    // ---- epilogue: C/D layout -> lane holds N=lsub, M = base + lhi*8 + r ----
#pragma unroll
    for (int i = 0; i < 4; ++i) {
        const int mbase = m0 + mw + 16 * i + lhi * 8;
#pragma unroll
        for (int j = 0; j < 2; ++j) {
            const int n  = n0 + nw + 16 * j + lsub;
            const float bv = bias[n];
#pragma unroll
            for (int r = 0; r < 8; ++r)
                out[(size_t)(mbase + r) * OUT_DIM + n] = acc[i][j][r] + bv;
        }
    }
}

extern "C" void kernel_launch(void* const* d_in, const int* in_sizes, int n_in,
                              void* d_out, int out_size, void* d_ws, size_t ws_size,
                              hipStream_t stream) {
    (void)in_sizes; (void)n_in; (void)out_size; (void)d_ws; (void)ws_size;
    const float* x   = (const float*)d_in[0];
    const float* W0  = (const float*)d_in[1];
    const float* b   = (const float*)d_in[2];
    const float* w1  = (const float*)d_in[3];
    const float* w2a = (const float*)d_in[4];
    const float* w2b = (const float*)d_in[5];
    float* out = (float*)d_out;

    dim3 grid(OUT_DIM / BN, M_TOT / BM);   // 32 x 128 blocks
    lokr_fused_wmma_gemm<<<grid, 256, 0, stream>>>(x, W0, b, w1, w2a, w2b, out);
}